// ODELayer_38594576122135
// MI455X (gfx1250) — compile-verified
//
#include <hip/hip_runtime.h>
#include <hip/hip_bf16.h>
#include <cstdint>

// ODE layer (LTC-style): per step t, per (b,u):
//   f = sigmoid(sigma*(u_in-mu));  3 unfolds of x += dt'(-x + S1 - x*S0)
// collapse to x <- c*x + d applied 3x, with
//   c = 1 - dt'(1+S0), d = dt'*S1, S0 = sum_din f, S1 = sum_din f*A.
// tanh form: f = 0.5*tanh(z)+0.5, z = 0.5*sigma*(in-mu). Folding the affine:
//   T0 = sum tanh(z), T1 = sum tanh(z)*A, Atot = sum A[u,:]
//   c = (1 - 33*dt') - (dt'/2)*T0 = -0.1 - (dt'/2)*T0
//   d = (dt'/2)*T1 + (dt'/2)*Atot
// Compute-bound: 4.3G transcendentals + ~13G VALU; memory floor ~16us.

#define B_   64
#define T_   8192
#define U_   128
#define DIN_ 64
#define CH   32            // timesteps per LDS chunk (8 KB)
#define NCH  (T_ / CH)

#if __has_builtin(__builtin_amdgcn_tanhf)
  #define HAS_TANH 1
  #define TANHF(x) __builtin_amdgcn_tanhf(x)
  #pragma message("CDNA5 PATH: v_tanh via __builtin_amdgcn_tanhf")
#elif __has_builtin(__builtin_amdgcn_tanh_f32)
  #define HAS_TANH 1
  #define TANHF(x) __builtin_amdgcn_tanh_f32(x)
  #pragma message("CDNA5 PATH: v_tanh via __builtin_amdgcn_tanh_f32")
#else
  #define HAS_TANH 0
  #pragma message("CDNA5 PATH: exp2+rcp fallback (no tanh builtin)")
#endif

// Async global -> LDS copy, 16B per lane (ASYNCcnt-tracked). LDS operand is
// the low 32 bits of the generic pointer (HW uses addr[31:0] as LDS offset).
__device__ __forceinline__ void async_ld16(const void* g, void* l) {
  unsigned loff = (unsigned)(uintptr_t)l;
  asm volatile("global_load_async_to_lds_b128 %0, %1, off"
               :: "v"(loff), "v"(g) : "memory");
}
__device__ __forceinline__ void wait_async0() {
  asm volatile("s_wait_asynccnt 0" ::: "memory");
}

__global__ __launch_bounds__(256) void ode_scan_kernel(
    const float* __restrict__ inp,    // [B, T, DIN]
    const float* __restrict__ A,      // [U, DIN]
    const float* __restrict__ sigma,  // [U, DIN]
    const float* __restrict__ mu,     // [U, DIN]
    const float* __restrict__ x0,     // [U]
    float* __restrict__ out)          // [B, T, U]
{
  __shared__ __align__(16) float smem[2][CH * DIN_];

  const int tid = threadIdx.x;
  const int b   = blockIdx.x >> 2;                             // 64 batches
  const int u0  = ((blockIdx.x & 3) << 5) + ((tid >> 5) << 2); // 4 units/wave
  const int su  = (tid >> 3) & 3;                              // sub-unit
  const int dl  = tid & 7;                                     // din-lane
  const int u   = u0 + su;

  const float dtu = 0.1f / 3.0f;   // DT / ODE_UNFOLDS

#if HAS_TANH
  const float SCALE = 0.5f;
#else
  const float SCALE = 1.4426950408889634f;  // log2(e)
#endif

  // Per-lane constants for this lane's 8 din values of unit u.
  float ksig[8], kmu[8], av[8];
  float asum = 0.0f;
  {
    const float* sp = sigma + u * DIN_ + dl * 8;
    const float* mp = mu    + u * DIN_ + dl * 8;
    const float* ap = A     + u * DIN_ + dl * 8;
#pragma unroll
    for (int k = 0; k < 8; ++k) {
      float s = sp[k] * SCALE;
      ksig[k] = s;
      kmu[k]  = s * mp[k];
      av[k]   = ap[k];
      asum   += ap[k];
    }
  }
#pragma unroll
  for (int m = 1; m < 8; m <<= 1) asum += __shfl_xor(asum, m);  // sum A[u,:]

  // Recurrence-coefficient constants (tanh path):
  //   c = K0 + K1*T0tot, d = K1h*T1tot + dconst
  const float K0  = 1.0f - 33.0f * dtu;   // = -0.1 exactly for dt'=0.1/3
  const float K1  = -0.5f * dtu;
  const float K1h =  0.5f * dtu;
  const float dconst = K1h * asum;
  const float cmu = 1.0f - dtu;           // used by fallback path

  float x = x0[u];

  const float* gin = inp + (size_t)b * T_ * DIN_;
  float* op = out + (size_t)b * T_ * U_ + u;

  // Prologue: stage chunk 0 (256 threads x 2 x 16B = 8 KB).
  {
    const char* g = (const char*)gin;
    char* l = (char*)&smem[0][0];
    async_ld16(g + tid * 16,        l + tid * 16);
    async_ld16(g + 4096 + tid * 16, l + 4096 + tid * 16);
    wait_async0();
    __syncthreads();
  }

  for (int c = 0; c < NCH; ++c) {
    // Prefetch next chunk into the other buffer while computing this one.
    if (c + 1 < NCH) {
      const char* g = (const char*)(gin + (size_t)(c + 1) * CH * DIN_);
      char* l = (char*)&smem[(c + 1) & 1][0];
      async_ld16(g + tid * 16,        l + tid * 16);
      async_ld16(g + 4096 + tid * 16, l + 4096 + tid * 16);
    }

    const float* lb = &smem[c & 1][0];
    float* opt = op + (size_t)c * CH * U_;

#pragma unroll 4
    for (int tl = 0; tl < CH; ++tl) {
      const float4* ip = (const float4*)(lb + tl * DIN_ + dl * 8);
      float4 v0 = ip[0];
      float4 v1 = ip[1];
      float in8[8] = {v0.x, v0.y, v0.z, v0.w, v1.x, v1.y, v1.z, v1.w};

#if HAS_TANH
      float T0 = 0.0f, T1 = 0.0f;
#pragma unroll
      for (int k = 0; k < 8; ++k) {
        float z  = __builtin_fmaf(ksig[k], in8[k], -kmu[k]);
        float th = TANHF(z);
        T0 += th;
        T1 = __builtin_fmaf(th, av[k], T1);
      }
#pragma unroll
      for (int m = 1; m < 8; m <<= 1) {
        T0 += __shfl_xor(T0, m);
        T1 += __shfl_xor(T1, m);
      }
      float cc = __builtin_fmaf(K1,  T0, K0);
      float dd = __builtin_fmaf(K1h, T1, dconst);
#else
      float S0 = 0.0f, S1 = 0.0f;
#pragma unroll
      for (int k = 0; k < 8; ++k) {
        float z = __builtin_fmaf(ksig[k], in8[k], -kmu[k]);
        float f = __builtin_amdgcn_rcpf(1.0f + __builtin_amdgcn_exp2f(-z));
        S0 += f;
        S1 = __builtin_fmaf(f, av[k], S1);
      }
#pragma unroll
      for (int m = 1; m < 8; m <<= 1) {
        S0 += __shfl_xor(S0, m);
        S1 += __shfl_xor(S1, m);
      }
      float cc = __builtin_fmaf(-dtu, S0, cmu);
      float dd = dtu * S1;
#endif
      x = __builtin_fmaf(cc, x, dd);   // 3 ODE unfolds
      x = __builtin_fmaf(cc, x, dd);
      x = __builtin_fmaf(cc, x, dd);

      if (dl == 0) opt[tl * U_] = x;   // 4 lanes -> 16 contiguous bytes
    }

    wait_async0();       // prefetched chunk has landed in LDS
    __syncthreads();     // everyone done reading before buffer reuse
  }
}

extern "C" void kernel_launch(void* const* d_in, const int* in_sizes, int n_in,
                              void* d_out, int out_size, void* d_ws, size_t ws_size,
                              hipStream_t stream) {
  (void)in_sizes; (void)n_in; (void)out_size; (void)d_ws; (void)ws_size;
  const float* inp   = (const float*)d_in[0];
  const float* A     = (const float*)d_in[1];
  const float* sigma = (const float*)d_in[2];
  const float* mu    = (const float*)d_in[3];
  const float* x0    = (const float*)d_in[4];
  float* out = (float*)d_out;

  dim3 grid(B_ * (U_ / 32));   // 256 blocks: (batch, 32-unit tile)
  dim3 block(256);             // 8 waves * wave32
  ode_scan_kernel<<<grid, block, 0, stream>>>(inp, A, sigma, mu, x0, out);
}